// CausalConv1d_60765197304309
// MI455X (gfx1250) — compile-verified
//
#include <hip/hip_runtime.h>
#include <cstdint>
#include <cstddef>

// ---------------------------------------------------------------------------
// Causal depthwise conv1d (K=4) over (T=8192, C=8448) fp32 for MI455X.
// Pure HBM-streaming problem: ~554 MB total @ 23.3 TB/s => ~24us floor.
// Data path: GLOBAL_LOAD_ASYNC_TO_LDS_B128 (ASYNCcnt) into wave-private LDS
// rings (no barriers), ds_load_b128 reads with immediate offsets (fully
// unrolled), FMA, non-temporal global_store_b128. Read amplification 68/64.
// ---------------------------------------------------------------------------

typedef float v4f __attribute__((ext_vector_type(4)));

#define C_DIM 8448
#define T_DIM 8192
#define K_TAPS 4
#define ROWS 64            // output rows per block
#define RING 16            // LDS ring slots per wave (rows)
#define WAVE_FLOATS 128    // 32 lanes * 4 floats
#define NWAVES 6           // 192 threads
#define SEG1_OFF 67108864ull            // 8192*8192
#define SEG2_OFF 68157440ull            // + 8192*128
#define STATE_OFF 69206016ull           // + 8192*128

#if __has_builtin(__builtin_amdgcn_s_wait_asynccnt)
#define WAIT_ASYNC(n) __builtin_amdgcn_s_wait_asynccnt(n)
#else
#define WAIT_ASYNC(n) asm volatile("s_wait_asynccnt %0" :: "n"(n) : "memory")
#endif

// One async DMA of 16B per lane: global (per-lane addr) -> LDS (per-lane addr).
// CDNA5: GLOBAL_LOAD_ASYNC_TO_LDS_B128, tracked by ASYNCcnt.
__device__ __forceinline__ void async_row_load(const float* gsrc, const float* lds_dst) {
    unsigned loff = (unsigned)(size_t)(const void*)lds_dst;          // low 32b = DS-space addr
    unsigned long long ga = (unsigned long long)(uintptr_t)gsrc;
    asm volatile("global_load_async_to_lds_b128 %0, %1, off"
                 :: "v"(loff), "v"(ga)
                 : "memory");
}

__global__ __launch_bounds__(192)
void conv1d_k4_kernel(const float* __restrict__ inputs,
                      const float* __restrict__ state,
                      const float* __restrict__ weight,
                      const float* __restrict__ bias,
                      float* __restrict__ out)
{
    __shared__ float lds[NWAVES * RING * WAVE_FLOATS];   // 48 KB

    const int tid  = (int)threadIdx.x;
    const int wave = tid >> 5;
    const int lane = tid & 31;
    const int c0   = ((int)blockIdx.x * 192 + tid) * 4;  // this thread's first channel
    const int t0   = (int)blockIdx.y * ROWS;

    float* wlds = &lds[wave * (RING * WAVE_FLOATS)];
    const int laneOff = lane * 4;

    // ring-slot address for padded-local row r (compile-time offset after unroll)
    auto sl = [&](int r) -> float* {
        return wlds + (((r & (RING - 1)) * WAVE_FLOATS) + laneOff);
    };

    // per-thread conv weights (weight is (C,4) row-major: one float4 per channel) + bias
    const v4f w0 = *(const v4f*)(weight + (size_t)(c0 + 0) * 4);
    const v4f w1 = *(const v4f*)(weight + (size_t)(c0 + 1) * 4);
    const v4f w2 = *(const v4f*)(weight + (size_t)(c0 + 2) * 4);
    const v4f w3 = *(const v4f*)(weight + (size_t)(c0 + 3) * 4);
    const v4f bv = *(const v4f*)(bias + c0);

    // output segment: channels [0,8192) stride 8192; [8192,8320) stride 128; [8320,8448) stride 128
    // (segment boundaries are 128-channel aligned => wave-uniform, stores stay 512B-contiguous)
    size_t obase; size_t ostride;
    if (c0 < 8192)      { obase = (size_t)c0;                 ostride = 8192; }
    else if (c0 < 8320) { obase = SEG1_OFF + (c0 - 8192);     ostride = 128;  }
    else                { obase = SEG2_OFF + (c0 - 8320);     ostride = 128;  }
    float* op = out + obase + (size_t)t0 * ostride;          // running output row pointer

    // ---- prologue: prime rows 0..11 (handles state rows only when t0 == 0) ----
    #pragma unroll
    for (int p = 0; p < 12; ++p) {
        const int P = t0 + p;                                // padded row index
        const float* src = (P < K_TAPS)
            ? (state  + (size_t)P * C_DIM + c0)
            : (inputs + (size_t)(P - K_TAPS) * C_DIM + c0);
        async_row_load(src, sl(p));
    }

    // compute 4 output rows of group g from ring rows 4g..4g+6 (all offsets fold)
    auto compute = [&](int g) {
        v4f p[7];
        #pragma unroll
        for (int i = 0; i < 7; ++i) p[i] = *(const v4f*)sl(4 * g + i);
        #pragma unroll
        for (int rr = 0; rr < 4; ++rr) {
            v4f acc = bv;
            acc.x += p[rr].x * w0.x + p[rr+1].x * w0.y + p[rr+2].x * w0.z + p[rr+3].x * w0.w;
            acc.y += p[rr].y * w1.x + p[rr+1].y * w1.y + p[rr+2].y * w1.z + p[rr+3].y * w1.w;
            acc.z += p[rr].z * w2.x + p[rr+1].z * w2.y + p[rr+2].z * w2.z + p[rr+3].z * w2.w;
            acc.w += p[rr].w * w3.x + p[rr+1].w * w3.y + p[rr+2].w * w3.z + p[rr+3].w * w3.w;
            __builtin_nontemporal_store(acc, (v4f*)op);
            op += ostride;
        }
    };

    // ---- steady state: rows >= 12 always come from `inputs`; running pointer ----
    const float* gp = inputs + (size_t)(t0 + 12 - K_TAPS) * C_DIM + c0;  // padded row 12

    #pragma unroll
    for (int g = 0; g < 14; ++g) {
        const int pb = 4 * (g + 3);                          // issue rows pb..pb+3
        async_row_load(gp,             sl(pb + 0));
        async_row_load(gp +     C_DIM, sl(pb + 1));
        async_row_load(gp + 2 * C_DIM, sl(pb + 2));
        async_row_load(gp + 3 * C_DIM, sl(pb + 3));
        gp += 4 * C_DIM;
        // issued = 4g+16 rows; need first 4g+7 complete -> outstanding <= 9
        WAIT_ASYNC(9);
        compute(g);
    }

    // ---- drain: groups 14,15 (need rows <= 66; all 68 issued) ----
    WAIT_ASYNC(0);
    compute(14);
    compute(15);
}

// updated_state = inputs[T-4 : T, :]  (4*8448 floats = 8448 float4, exact grid)
__global__ __launch_bounds__(256)
void state_copy_kernel(const float* __restrict__ inputs, float* __restrict__ out_state)
{
    const int i = (int)blockIdx.x * 256 + (int)threadIdx.x;   // float4 index, 0..8447
    const v4f* src = (const v4f*)(inputs + (size_t)(T_DIM - K_TAPS) * C_DIM);
    v4f v = src[i];
    __builtin_nontemporal_store(v, ((v4f*)out_state) + i);
}

extern "C" void kernel_launch(void* const* d_in, const int* in_sizes, int n_in,
                              void* d_out, int out_size, void* d_ws, size_t ws_size,
                              hipStream_t stream) {
    const float* inputs = (const float*)d_in[0];   // (8192, 8448)
    const float* state  = (const float*)d_in[1];   // (4, 8448)
    const float* weight = (const float*)d_in[2];   // (8448, 4)
    const float* bias   = (const float*)d_in[3];   // (8448,)
    float* out = (float*)d_out;

    dim3 grid(11, 128);   // 11 channel tiles * 128 time tiles
    dim3 block(192);      // 6 waves
    conv1d_k4_kernel<<<grid, block, 0, stream>>>(inputs, state, weight, bias, out);
    state_copy_kernel<<<33, 256, 0, stream>>>(inputs, out + STATE_OFF);
}